// ReprojectionLayer_10660108828790
// MI455X (gfx1250) — compile-verified
//
#include <hip/hip_runtime.h>
#include <hip/hip_bf16.h>

typedef __attribute__((ext_vector_type(2))) float v2f;
typedef __attribute__((ext_vector_type(8))) float v8f;

#define GSZ    100
#define G3     1000000
#define GROUPS_PER_B 62500     /* G3 / 16 points-per-WMMA-group */
#define BB     2
#define CC     8
#define JJ     12
#define IMGW   1280
#define IMGH   1024
#define HMW    (IMGW/2)
#define HMH    (IMGH/2)
#define HWSZ   (HMW*HMH)

// One wave projects 16 grid points through all 8 cameras with two
// V_WMMA_F32_16X16X4_F32 ops (4 cameras per WMMA, camera padded to 4 rows),
// then each lane gathers 12 joints for its 4 cameras and a shfl_xor(16)
// reduces the two camera halves.
__global__ __launch_bounds__(256)
void reproj_wmma_kernel(const float* __restrict__ hm,      // [B,C,J,HMH,HMW]
                        const float* __restrict__ center,  // [B,3]
                        const float* __restrict__ Mmat,    // [B,C,4,3]
                        const float* __restrict__ intr,    // [B,C,3,3]
                        const float* __restrict__ dist,    // [B,C,1,5]
                        float* __restrict__ out)           // [B,J,G,G,G]
{
    const int lane   = threadIdx.x & 31;
    const int widx   = threadIdx.x >> 5;
    const int wave   = blockIdx.x * (blockDim.x >> 5) + widx;
    const int nwaves = gridDim.x * (blockDim.x >> 5);

    const int lh   = lane >> 4;   // half-wave select (D rows 0-7 vs 8-15)
    const int m    = lane & 15;   // A row / B,D column
    const int caml = m >> 2;      // camera-in-group (rows padded to 4/cam)
    const int ch   = m & 3;       // 0..2 = px,py,pz ; 3 = pad row
    const int k0   = lh * 2;      // A K-index for vgpr0 (layout: K = v + 2*lh)

    for (int b = 0; b < BB; ++b) {
        const float cenx = center[b*3+0];
        const float ceny = center[b*3+1];
        const float cenz = center[b*3+2];

        // A fragments: A[m,k] = M[b, cam(m), k, ch(m)] (pad rows = 0)
        v2f a0, a1;
        {
            const float* M0 = Mmat + ((size_t)(b*CC + caml) * 4) * 3;
            const float* M1 = Mmat + ((size_t)(b*CC + 4 + caml) * 4) * 3;
            a0.x = (ch < 3) ? M0[(k0    )*3 + ch] : 0.f;
            a0.y = (ch < 3) ? M0[(k0 + 1)*3 + ch] : 0.f;
            a1.x = (ch < 3) ? M1[(k0    )*3 + ch] : 0.f;
            a1.y = (ch < 3) ? M1[(k0 + 1)*3 + ch] : 0.f;
        }

        // this lane's 4 cameras: lanes 0-15 -> {0,1,4,5}; lanes 16-31 -> {2,3,6,7}
        int cams[4];
        cams[0] = lh*2;     cams[1] = lh*2 + 1;
        cams[2] = lh*2 + 4; cams[3] = lh*2 + 5;

        float fx[4], fy[4], cx[4], cy[4], rfx[4], rfy[4], dk1[4], dk2[4];
        const float* hbase[4];
        #pragma unroll
        for (int i = 0; i < 4; ++i) {
            const int c = cams[i];
            const float* ip = intr + (size_t)(b*CC + c) * 9;
            fx[i] = ip[0];  fy[i] = ip[4];   // [0][0], [1][1]
            cx[i] = ip[6];  cy[i] = ip[7];   // [2][0], [2][1]
            rfx[i] = 1.0f / fx[i];
            rfy[i] = 1.0f / fy[i];
            const float* dp = dist + (size_t)(b*CC + c) * 5;
            dk1[i] = dp[0]; dk2[i] = dp[1];
            hbase[i] = hm + (size_t)(b*CC + c) * JJ * HWSZ;
        }

        for (int grp = wave; grp < GROUPS_PER_B; grp += nwaves) {
            const int q = grp * 16 + m;                 // point index (n)
            int ix = q / 10000;
            int r  = q - ix * 10000;
            int iy = r / 100;
            int iz = r - iy * 100;
            float gx = (float)(ix - 50) * 2.0f + cenx;
            float gy = (float)(iy - 50) * 2.0f + ceny;
            float gz = (float)(iz - 50) * 2.0f + cenz;

            // B fragment (4x16): vgpr v, half lh holds K = v + 2*lh
            v2f bv;
            bv.x = lh ? gz : gx;      // K=2 : K=0
            bv.y = lh ? 1.0f : gy;    // K=3 : K=1

            v8f cz = {0.f,0.f,0.f,0.f,0.f,0.f,0.f,0.f};
            v8f d0 = __builtin_amdgcn_wmma_f32_16x16x4_f32(
                false, a0, false, bv, (short)0, cz, false, false);
            v8f d1 = __builtin_amdgcn_wmma_f32_16x16x4_f32(
                false, a1, false, bv, (short)0, cz, false, false);

            float acc[JJ];
            #pragma unroll
            for (int j = 0; j < JJ; ++j) acc[j] = 0.f;

            #pragma unroll
            for (int i = 0; i < 4; ++i) {
                float pjx, pjy, pjz;
                if      (i == 0) { pjx = d0[0]; pjy = d0[1]; pjz = d0[2]; }
                else if (i == 1) { pjx = d0[4]; pjy = d0[5]; pjz = d0[6]; }
                else if (i == 2) { pjx = d1[0]; pjy = d1[1]; pjz = d1[2]; }
                else             { pjx = d1[4]; pjy = d1[5]; pjz = d1[6]; }

                float invz = __builtin_amdgcn_rcpf(pjz);   // z ~ 1000, rcp safe
                float px = pjx * invz;
                float py = pjy * invz;
                float xn = (px - cx[i]) * rfx[i];
                float yn = (py - cy[i]) * rfy[i];
                float r2  = xn*xn + yn*yn;
                float rad = 1.0f + dk1[i]*r2 + dk2[i]*r2*r2;
                float xd = fminf(fmaxf(xn*rad*fx[i] + cx[i], 0.f), (float)(IMGW-1));
                float yd = fminf(fmaxf(yn*rad*fy[i] + cy[i], 0.f), (float)(IMGH-1));
                int idx = (int)(yd * 0.5f) * HMW + (int)(xd * 0.5f);

                const float* hp = hbase[i] + idx;
                #pragma unroll
                for (int j = 0; j < JJ; ++j)
                    acc[j] += hp[(size_t)j * HWSZ];        // 12 joint gathers
            }

            // combine the two camera halves (lane <-> lane^16)
            #pragma unroll
            for (int j = 0; j < JJ; ++j)
                acc[j] += __shfl_xor(acc[j], 16, 32);

            if (lane < 16) {
                float* op = out + (size_t)b * JJ * G3 + q;
                #pragma unroll
                for (int j = 0; j < JJ; ++j)
                    op[(size_t)j * G3] = acc[j] * 0.125f;  // mean over 8 cams
            }
        }
    }
}

extern "C" void kernel_launch(void* const* d_in, const int* in_sizes, int n_in,
                              void* d_out, int out_size, void* d_ws, size_t ws_size,
                              hipStream_t stream) {
    const float* hm     = (const float*)d_in[0];
    const float* center = (const float*)d_in[1];
    const float* Mmat   = (const float*)d_in[2];
    const float* intr   = (const float*)d_in[3];
    const float* dist   = (const float*)d_in[4];
    float* out = (float*)d_out;

    dim3 block(256);       // 8 waves/block (wave32)
    dim3 grid(1024);       // 8192 waves, grid-stride over 62500 groups per batch
    hipLaunchKernelGGL(reproj_wmma_kernel, grid, block, 0, stream,
                       hm, center, Mmat, intr, dist, out);
}